// GCNLayerChunked_24790551232877
// MI455X (gfx1250) — compile-verified
//
#include <hip/hip_runtime.h>
#include <hip/hip_bf16.h>

typedef __attribute__((ext_vector_type(2))) float v2f;
typedef __attribute__((ext_vector_type(8))) float v8f;

#define F 128            // F_IN == F_OUT == 128
#define GEMM_BLOCK 256   // 8 waves; each wave owns one 16-wide column tile

// ---------------------------------------------------------------------------
// Kernel 1: h = x @ W^T + b   via V_WMMA_F32_16X16X4_F32
// One block = 16 rows of x, 8 waves cover n = 0..127 (16 cols each).
// A-matrix (16x4 f32): lane&15 -> M row; lane>>4 selects K pair {k,k+1} vs {k+2,k+3}
// B-matrix (4x16 f32): lane&15 -> N col; same K-pair split. B[k][n] = W[n][k],
// so both operands are contiguous float2 loads from row-major x / W.
// ---------------------------------------------------------------------------
__global__ __launch_bounds__(GEMM_BLOCK)
void gcn_gemm_wmma_f32(const float* __restrict__ x,
                       const float* __restrict__ W,
                       const float* __restrict__ b,
                       float* __restrict__ h,
                       int N)
{
    const int lane = threadIdx.x & 31;
    const int wave = threadIdx.x >> 5;        // 0..7 -> column tile
    const int m0   = blockIdx.x * 16;
    const int n0   = wave * 16;
    const int l15  = lane & 15;
    const int half = lane >> 4;               // 0 or 1 (K pair select)

    // Clamp row for loads (N=100000 is a multiple of 16, but stay safe).
    int mrow = m0 + l15; if (mrow >= N) mrow = N - 1;

    const float* xrow = x + (size_t)mrow        * F + 2 * half;
    const float* wrow = W + (size_t)(n0 + l15)  * F + 2 * half;

    // Bias: C[v] lane layout is N = lane&15 for every accumulator VGPR.
    const float bb = b[n0 + l15];
    v8f c = { bb, bb, bb, bb, bb, bb, bb, bb };

    #pragma unroll
    for (int k = 0; k < F; k += 4) {
        v2f a  = *(const v2f*)(xrow + k);
        v2f bm = *(const v2f*)(wrow + k);
        // 8 args: (neg_a, A, neg_b, B, c_mod, C, reuse_a, reuse_b)
        c = __builtin_amdgcn_wmma_f32_16x16x4_f32(
                false, a, false, bm, (short)0, c, false, false);
    }

    // D layout: VGPR v, lanes 0-15 -> M=v, N=lane; lanes 16-31 -> M=v+8.
    float* hout = h + (size_t)m0 * F + n0;
    #pragma unroll
    for (int v = 0; v < 8; ++v) {
        const int m = m0 + v + 8 * half;
        if (m < N) hout[(size_t)(v + 8 * half) * F + l15] = c[v];
    }
}

// ---------------------------------------------------------------------------
// Kernel 2: zero the output (harness poisons d_out with 0xAA).
// ---------------------------------------------------------------------------
__global__ void gcn_zero_out(float4* __restrict__ out, int nvec)
{
    const int i = blockIdx.x * blockDim.x + threadIdx.x;
    if (i < nvec) out[i] = make_float4(0.f, 0.f, 0.f, 0.f);
}

// ---------------------------------------------------------------------------
// Kernel 3: out[dst] += w * h[src].  One wave32 per edge; each lane covers 4
// contiguous features: 16B coalesced gather + 4 hardware f32 atomics into L2.
// h + out together (~102 MB) fit in the 192 MB L2, so this phase is
// L2-resident after first touch.
// ---------------------------------------------------------------------------
__global__ __launch_bounds__(256)
void gcn_edge_scatter(const float* __restrict__ h,
                      const int*   __restrict__ src,
                      const int*   __restrict__ dst,
                      const float* __restrict__ ew,
                      float* __restrict__ out,
                      int E)
{
    const int lane = threadIdx.x & 31;
    const int e    = blockIdx.x * 8 + (threadIdx.x >> 5);   // 8 waves/block
    if (e >= E) return;

    // Edge data is wave-uniform: pin it in SGPRs for scalar addressing.
    const int   s  = __builtin_amdgcn_readfirstlane(src[e]);
    const int   d  = __builtin_amdgcn_readfirstlane(dst[e]);
    const float we = ew[e];

    const float4 v = *(const float4*)(h + (size_t)s * F + lane * 4);
    float* o = out + (size_t)d * F + lane * 4;

    unsafeAtomicAdd(o + 0, v.x * we);
    unsafeAtomicAdd(o + 1, v.y * we);
    unsafeAtomicAdd(o + 2, v.z * we);
    unsafeAtomicAdd(o + 3, v.w * we);
}

// ---------------------------------------------------------------------------
// Launcher. Inputs (setup_inputs order): x, src_idx, dst_idx, edge_weight, W, b
// Workspace: h = [N, 128] fp32 (51.2 MB).
// ---------------------------------------------------------------------------
extern "C" void kernel_launch(void* const* d_in, const int* in_sizes, int n_in,
                              void* d_out, int out_size, void* d_ws, size_t ws_size,
                              hipStream_t stream)
{
    const float* x   = (const float*)d_in[0];
    const int*   src = (const int*)  d_in[1];
    const int*   dst = (const int*)  d_in[2];
    const float* ew  = (const float*)d_in[3];
    const float* W   = (const float*)d_in[4];
    const float* b   = (const float*)d_in[5];
    float*       out = (float*)d_out;
    float*       h   = (float*)d_ws;

    const int N = in_sizes[0] / F;   // 100000
    const int E = in_sizes[1];       // 600000

    // 1) Linear projection with WMMA.
    const int mtiles = (N + 15) / 16;
    gcn_gemm_wmma_f32<<<mtiles, GEMM_BLOCK, 0, stream>>>(x, W, b, h, N);

    // 2) Zero output.
    const int nvec = out_size / 4;   // out_size = N*128, divisible by 4
    gcn_zero_out<<<(nvec + 255) / 256, 256, 0, stream>>>((float4*)out, nvec);

    // 3) Weighted gather / scatter-add.
    const int nblocks = (E + 7) / 8;
    gcn_edge_scatter<<<nblocks, 256, 0, stream>>>(h, src, dst, ew, out, E);
}